// GNNClassifier_15075335209588
// MI455X (gfx1250) — compile-verified
//
#include <hip/hip_runtime.h>
#include <hip/hip_bf16.h>
#include <math.h>

#define NUM_GRAPHS 256

typedef __attribute__((ext_vector_type(2))) float v2f;
typedef __attribute__((ext_vector_type(8))) float v8f;

// ---------------------------------------------------------------- utilities
__global__ __launch_bounds__(256) void k_fill(float* __restrict__ p, float v, long long n) {
    long long i = (long long)blockIdx.x * blockDim.x + threadIdx.x;
    if (i < n) p[i] = v;
}

__global__ __launch_bounds__(256) void k_deg_count(const long long* __restrict__ col,
                                                   float* __restrict__ deg, long long E) {
    long long i = (long long)blockIdx.x * blockDim.x + threadIdx.x;
    if (i < E) unsafeAtomicAdd(&deg[(int)col[i]], 1.0f);
}

__global__ __launch_bounds__(256) void k_rsqrt(float* __restrict__ p, long long n) {
    long long i = (long long)blockIdx.x * blockDim.x + threadIdx.x;
    if (i < n) p[i] = rsqrtf(p[i]);   // deg >= 1 always (self-loop)
}

// ------------------------------------------------- dense transform: h = A @ W
// A: [N,64] row-major, W: [64,64] row-major (k-major), out: [N,64].
// One wave computes a 16x64 tile with V_WMMA_F32_16X16X4_F32 (exact fp32).
template <bool RELU_IN>
__global__ __launch_bounds__(256) void k_gemm64(const float* __restrict__ A,
                                                const float* __restrict__ W,
                                                float* __restrict__ out, int N) {
    __shared__ float lw[64 * 64];
    for (int i = threadIdx.x; i < 64 * 64; i += 256) lw[i] = W[i];
    __syncthreads();

    const int lane  = threadIdx.x & 31;
    const int wave  = threadIdx.x >> 5;
    const int m0    = (blockIdx.x * 8 + wave) * 16;
    const int nlane = lane & 15;
    const int khalf = (lane >> 4) << 1;          // 0 or 2 (K sub-offset per half-wave)

    // A fragment row (clamped so EXEC stays all-1s; garbage rows masked at store)
    const int mrow = m0 + (lane & 15);
    const int arow = (mrow < N) ? mrow : (N - 1);
    const float* aptr = A + (size_t)arow * 64;

    v8f acc[4];
    acc[0] = {}; acc[1] = {}; acc[2] = {}; acc[3] = {};

#pragma unroll
    for (int kt = 0; kt < 16; ++kt) {
        const int kb = kt * 4 + khalf;
        float ax = aptr[kb + 0];
        float ay = aptr[kb + 1];
        if (RELU_IN) { ax = fmaxf(ax, 0.0f); ay = fmaxf(ay, 0.0f); }
        v2f a; a.x = ax; a.y = ay;
#pragma unroll
        for (int nt = 0; nt < 4; ++nt) {
            v2f b;
            b.x = lw[(kb + 0) * 64 + nt * 16 + nlane];
            b.y = lw[(kb + 1) * 64 + nt * 16 + nlane];
            acc[nt] = __builtin_amdgcn_wmma_f32_16x16x4_f32(
                false, a, false, b, (short)0, acc[nt], false, false);
        }
    }

    // C/D layout: VGPR r, lanes0-15 -> M=m0+r, lanes16-31 -> M=m0+8+r
    const int rbase = m0 + ((lane >> 4) << 3);
#pragma unroll
    for (int r = 0; r < 8; ++r) {
        const int row = rbase + r;
        if (row < N) {
            float* o = out + (size_t)row * 64 + nlane;
            o[0]  = acc[0][r];
            o[16] = acc[1][r];
            o[32] = acc[2][r];
            o[48] = acc[3][r];
        }
    }
}

// ---------------------------------------------- aggregation (gcn propagate)
// out[i] = bias + dis[i]^2 * h[i]   (self-loop message, norm = 1/deg)
__global__ __launch_bounds__(256) void k_agg_init(const float* __restrict__ h,
                                                  const float* __restrict__ dis,
                                                  const float* __restrict__ bias,
                                                  float* __restrict__ out, long long N) {
    long long idx = (long long)blockIdx.x * blockDim.x + threadIdx.x;
    if (idx >= N * 64) return;
    const int node = (int)(idx >> 6);
    const int f    = (int)(idx & 63);
    const float d  = dis[node];
    out[idx] = bias[f] + d * d * h[idx];
}

// out[col] += dis[row]*dis[col] * h[row]; 16 threads per edge, float4 each.
__global__ __launch_bounds__(256) void k_agg_edges(const float* __restrict__ h,
                                                   const long long* __restrict__ rows,
                                                   const long long* __restrict__ cols,
                                                   const float* __restrict__ dis,
                                                   float* __restrict__ out, long long E) {
    long long tid = (long long)blockIdx.x * blockDim.x + threadIdx.x;
    long long e = tid >> 4;
    if (e >= E) return;
    const int sub = (int)(tid & 15);
    const int r = (int)rows[e];
    const int c = (int)cols[e];
    const float w = dis[r] * dis[c];
    const float4 v = ((const float4*)(h + (size_t)r * 64))[sub];
    float* o = out + (size_t)c * 64 + sub * 4;
    unsafeAtomicAdd(o + 0, w * v.x);
    unsafeAtomicAdd(o + 1, w * v.y);
    unsafeAtomicAdd(o + 2, w * v.z);
    unsafeAtomicAdd(o + 3, w * v.w);
}

// ----------------------------------------------------------------- pooling
// sums[batch[i]] += relu(h[i]); cnt[batch[i]] += 1
__global__ __launch_bounds__(256) void k_pool(const float* __restrict__ h,
                                              const long long* __restrict__ batch,
                                              float* __restrict__ sums,
                                              float* __restrict__ cnt, long long N) {
    long long tid = (long long)blockIdx.x * blockDim.x + threadIdx.x;
    long long node = tid >> 4;
    if (node >= N) return;
    const int sub = (int)(tid & 15);
    const int g = (int)batch[node];
    const float4 v = ((const float4*)(h + (size_t)node * 64))[sub];
    float* o = sums + (size_t)g * 64 + sub * 4;
    unsafeAtomicAdd(o + 0, fmaxf(v.x, 0.0f));
    unsafeAtomicAdd(o + 1, fmaxf(v.y, 0.0f));
    unsafeAtomicAdd(o + 2, fmaxf(v.z, 0.0f));
    unsafeAtomicAdd(o + 3, fmaxf(v.w, 0.0f));
    if (sub == 0) unsafeAtomicAdd(&cnt[g], 1.0f);
}

// -------------------------------------------------------------- MLP head
// mean -> Linear(64,32)+ReLU -> Linear(32,2) -> log_softmax. One thread/graph.
__global__ __launch_bounds__(256) void k_head(const float* __restrict__ sums,
                                              const float* __restrict__ cnt,
                                              const float* __restrict__ Wl1,
                                              const float* __restrict__ bl1,
                                              const float* __restrict__ Wl2,
                                              const float* __restrict__ bl2,
                                              float* __restrict__ out) {
    __shared__ float sW1[64 * 32];
    __shared__ float sW2[32 * 2];
    __shared__ float sb1[32];
    __shared__ float sb2[2];
    for (int i = threadIdx.x; i < 64 * 32; i += 256) sW1[i] = Wl1[i];
    if (threadIdx.x < 64) sW2[threadIdx.x] = Wl2[threadIdx.x];
    if (threadIdx.x < 32) sb1[threadIdx.x] = bl1[threadIdx.x];
    if (threadIdx.x < 2)  sb2[threadIdx.x] = bl2[threadIdx.x];
    __syncthreads();

    const int g = threadIdx.x;          // 256 graphs, 256 threads
    const float inv = 1.0f / fmaxf(cnt[g], 1.0f);

    float l0 = sb2[0], l1 = sb2[1];
#pragma unroll 4
    for (int j = 0; j < 32; ++j) {
        float a = sb1[j];
        for (int k = 0; k < 64; ++k)
            a += sums[g * 64 + k] * inv * sW1[k * 32 + j];
        a = fmaxf(a, 0.0f);
        l0 += a * sW2[j * 2 + 0];
        l1 += a * sW2[j * 2 + 1];
    }
    const float m = fmaxf(l0, l1);
    const float lse = m + logf(expf(l0 - m) + expf(l1 - m));
    out[g * 2 + 0] = l0 - lse;
    out[g * 2 + 1] = l1 - lse;
}

// ---------------------------------------------------------------- launcher
extern "C" void kernel_launch(void* const* d_in, const int* in_sizes, int n_in,
                              void* d_out, int out_size, void* d_ws, size_t ws_size,
                              hipStream_t stream) {
    const float*      x     = (const float*)d_in[0];
    const long long*  ei    = (const long long*)d_in[1];
    const long long*  batch = (const long long*)d_in[2];
    const float*      W1    = (const float*)d_in[3];
    const float*      b1    = (const float*)d_in[4];
    const float*      W2    = (const float*)d_in[5];
    const float*      b2    = (const float*)d_in[6];
    const float*      Wl1   = (const float*)d_in[7];
    const float*      bl1   = (const float*)d_in[8];
    const float*      Wl2   = (const float*)d_in[9];
    const float*      bl2   = (const float*)d_in[10];

    const long long N = in_sizes[0] / 64;
    const long long E = in_sizes[1] / 2;
    const long long* rows = ei;
    const long long* cols = ei + E;

    char* ws = (char*)d_ws;
    size_t off = 0;
    auto take = [&](size_t bytes) -> void* {
        void* p = ws + off;
        off += (bytes + 255) & ~(size_t)255;
        return p;
    };
    float* dis  = (float*)take((size_t)N * 4);             // deg -> dis in place
    float* buf0 = (float*)take((size_t)N * 64 * 4);        // x@W / h@W
    float* buf1 = (float*)take((size_t)N * 64 * 4);        // aggregated h
    float* sums = (float*)take((size_t)NUM_GRAPHS * 64 * 4);
    float* cnt  = (float*)take((size_t)NUM_GRAPHS * 4);

    const int T = 256;
    const unsigned gN   = (unsigned)((N + T - 1) / T);
    const unsigned gE   = (unsigned)((E + T - 1) / T);
    const unsigned gNF  = (unsigned)((N * 64 + T - 1) / T);
    const unsigned gE16 = (unsigned)((E * 16 + T - 1) / T);
    const unsigned gN16 = (unsigned)((N * 16 + T - 1) / T);
    const unsigned gGemm = (unsigned)((N + 127) / 128);    // 8 waves x 16 rows

    // symmetric gcn norm
    k_fill<<<gN, T, 0, stream>>>(dis, 1.0f, N);            // self-loop degree
    k_deg_count<<<gE, T, 0, stream>>>(cols, dis, E);
    k_rsqrt<<<gN, T, 0, stream>>>(dis, N);

    // layer 1: transform (WMMA) then propagate
    k_gemm64<false><<<gGemm, 256, 0, stream>>>(x, W1, buf0, (int)N);
    k_agg_init<<<gNF, T, 0, stream>>>(buf0, dis, b1, buf1, N);
    k_agg_edges<<<gE16, T, 0, stream>>>(buf0, rows, cols, dis, buf1, E);

    // layer 2: relu fused into GEMM loads
    k_gemm64<true><<<gGemm, 256, 0, stream>>>(buf1, W2, buf0, (int)N);
    k_agg_init<<<gNF, T, 0, stream>>>(buf0, dis, b2, buf1, N);
    k_agg_edges<<<gE16, T, 0, stream>>>(buf0, rows, cols, dis, buf1, E);

    // pool (relu fused into pooling loads) + head
    k_fill<<<(NUM_GRAPHS * 64 + T - 1) / T, T, 0, stream>>>(sums, 0.0f, NUM_GRAPHS * 64);
    k_fill<<<1, T, 0, stream>>>(cnt, 0.0f, NUM_GRAPHS);
    k_pool<<<gN16, T, 0, stream>>>(buf1, batch, sums, cnt, N);
    k_head<<<1, 256, 0, stream>>>(sums, cnt, Wl1, bl1, Wl2, bl2, (float*)d_out);
}